// SAGEConv_7224134992220
// MI455X (gfx1250) — compile-verified
//
#include <hip/hip_runtime.h>
#include <hip/hip_bf16.h>

#define D 128   // D_IN == D_OUT == 128

typedef __attribute__((ext_vector_type(2))) float v2f;
typedef __attribute__((ext_vector_type(8))) float v8f;

// ---------------------------------------------------------------------------
// Kernel 1: zero workspace (summed [N*D] and deg [N])
// ---------------------------------------------------------------------------
__global__ void sage_zero(float* __restrict__ p, int n) {
    int i = blockIdx.x * blockDim.x + threadIdx.x;
    if (i < n) p[i] = 0.0f;
}

// ---------------------------------------------------------------------------
// Kernel 2: degree counts (segment_sum of ones over dst)
// ---------------------------------------------------------------------------
__global__ void sage_deg(const int* __restrict__ dst, float* __restrict__ deg, int E) {
    int e = blockIdx.x * blockDim.x + threadIdx.x;
    if (e < E) atomicAdd(&deg[dst[e]], 1.0f);
}

// ---------------------------------------------------------------------------
// Kernel 3: edge scatter-add: summed[dst] += feat[src]
// 32 lanes per edge, each lane handles 4 consecutive floats (float4 load).
// ---------------------------------------------------------------------------
__global__ void sage_scatter(const float* __restrict__ feat,
                             const int* __restrict__ src,
                             const int* __restrict__ dst,
                             float* __restrict__ summed, int E) {
    int t    = blockIdx.x * blockDim.x + threadIdx.x;
    int e    = t >> 5;
    int lane = t & 31;
    if (e >= E) return;
    int s = src[e];
    int d = dst[e];
    const float4 v = *(const float4*)(feat + (size_t)s * D + lane * 4);
    float* o = summed + (size_t)d * D + lane * 4;
    atomicAdd(o + 0, v.x);
    atomicAdd(o + 1, v.y);
    atomicAdd(o + 2, v.z);
    atomicAdd(o + 3, v.w);
}

// ---------------------------------------------------------------------------
// Kernel 4: fused dual GEMM + bias via V_WMMA_F32_16X16X4_F32
//   out = (summed/max(deg,1)) @ Wn^T + feat @ Ws^T + b
// One block = 16-node row tile (N = 3125*16 exactly). 8 waves/block, each
// wave computes one 16x16 output tile (column tile n0 = wave*16).
// ---------------------------------------------------------------------------
__global__ void __launch_bounds__(256) sage_gemm(
        const float* __restrict__ feat,
        const float* __restrict__ Wn,     // [128 x 128], row = out feature
        const float* __restrict__ Ws,     // [128 x 128]
        const float* __restrict__ bias,   // [128]
        const float* __restrict__ summed, // [N x 128]
        const float* __restrict__ deg,    // [N]
        float* __restrict__ out) {        // [N x 128]
    __shared__ float hn_lds[16 * D];
    __shared__ float ft_lds[16 * D];

    const int tile = blockIdx.x;
    const int tid  = threadIdx.x;

    // Stage h_neigh (= summed * 1/max(deg,1)) and feat tiles into LDS.
    for (int idx = tid; idx < 16 * D; idx += 256) {
        int r    = idx >> 7;           // row within tile
        int k    = idx & (D - 1);      // feature index
        int node = tile * 16 + r;
        float dg  = deg[node];
        float inv = 1.0f / fmaxf(dg, 1.0f);
        hn_lds[idx] = summed[(size_t)node * D + k] * inv;
        ft_lds[idx] = feat[(size_t)node * D + k];
    }
    __syncthreads();

    const int lane = tid & 31;
    const int wid  = tid >> 5;
    const int n0   = wid * 16;          // output column tile base
    const int r    = lane & 15;         // A: M row / B: N col within tile
    const int half = lane >> 4;         // lanes 16-31 hold K+2 (A/B) or M+8 (C/D)
    const int kb   = half * 2;          // K sub-offset for A/B operands

    const float* wn_row = Wn + (size_t)(n0 + r) * D;  // B[k][n] = W[n][k]
    const float* ws_row = Ws + (size_t)(n0 + r) * D;

    // Two independent accumulator chains (better WMMA pipelining);
    // bias folded into one of them (C/D: column = n0 + r for every VGPR).
    float bval = bias[n0 + r];
    v8f acc_n;
    v8f acc_s;
#pragma unroll
    for (int i = 0; i < 8; ++i) { acc_n[i] = 0.0f; acc_s[i] = bval; }

#pragma unroll 4
    for (int kk = 0; kk < D; kk += 4) {
        // A tile operand: lane holds row r, K = kk+kb, kk+kb+1
        v2f a0 = *(const v2f*)(&hn_lds[r * D + kk + kb]);
        v2f b0 = *(const v2f*)(&wn_row[kk + kb]);
        acc_n = __builtin_amdgcn_wmma_f32_16x16x4_f32(
            false, a0, false, b0, (short)0, acc_n, false, false);

        v2f a1 = *(const v2f*)(&ft_lds[r * D + kk + kb]);
        v2f b1 = *(const v2f*)(&ws_row[kk + kb]);
        acc_s = __builtin_amdgcn_wmma_f32_16x16x4_f32(
            false, a1, false, b1, (short)0, acc_s, false, false);
    }

    // C/D layout: VGPR v -> M = v + 8*half, N = lane&15 (= r).
#pragma unroll
    for (int v = 0; v < 8; ++v) {
        out[(size_t)(tile * 16 + v + half * 8) * D + n0 + r] = acc_n[v] + acc_s[v];
    }
}

// ---------------------------------------------------------------------------
// Host launcher
// ---------------------------------------------------------------------------
extern "C" void kernel_launch(void* const* d_in, const int* in_sizes, int n_in,
                              void* d_out, int out_size, void* d_ws, size_t ws_size,
                              hipStream_t stream) {
    const float* feat = (const float*)d_in[0];   // [N, 128]
    const float* Wn   = (const float*)d_in[1];   // [128, 128]
    const float* Ws   = (const float*)d_in[2];   // [128, 128]
    const float* bias = (const float*)d_in[3];   // [128]
    const int*   src  = (const int*)d_in[4];     // [E]
    const int*   dst  = (const int*)d_in[5];     // [E]
    float*       out  = (float*)d_out;           // [N, 128]

    const int N = in_sizes[0] / D;               // 50000
    const int E = in_sizes[4];                   // 800000

    float* summed = (float*)d_ws;                // [N, 128]
    float* deg    = summed + (size_t)N * D;      // [N]

    // 1) zero workspace (summed + deg are contiguous)
    int zn = N * D + N;
    sage_zero<<<(zn + 255) / 256, 256, 0, stream>>>(summed, zn);

    // 2) degree counts
    sage_deg<<<(E + 255) / 256, 256, 0, stream>>>(dst, deg, E);

    // 3) edge scatter-add (32 lanes per edge)
    long long st = (long long)E * 32;
    sage_scatter<<<(unsigned)((st + 255) / 256), 256, 0, stream>>>(feat, src, dst, summed, E);

    // 4) fused dual WMMA GEMM + bias  (N is an exact multiple of 16)
    sage_gemm<<<N / 16, 256, 0, stream>>>(feat, Wn, Ws, bias, summed, deg, out);
}